// ChannelDeconv_20023137534460
// MI455X (gfx1250) — compile-verified
//
#include <hip/hip_runtime.h>

typedef __attribute__((ext_vector_type(2))) float v2f;
typedef __attribute__((ext_vector_type(8))) float v8f;

#define NCH    64
#define HW     16384      // 128*128
#define CHW    1048576    // 64*HW
#define NCOLS  524288     // 32*HW
#define NSUB   58255      // ceil(524288/9)
#define EPSI   0.01f

// workspace layout (floats)
#define WS_SUM   0        // 64 channel sums
#define WS_G     64       // 64x64 Gram accumulator
#define WS_D     4160     // 64x64 deconv matrix
#define WS_BIAS  8256     // 64 bias = -(D @ mean)
#define WS_ZERO  4160     // floats to zero each launch (sums + Gram)

__device__ __forceinline__ v8f wmma_f32x4(v2f a, v2f b, v8f c) {
  // D(16x16) = A(16x4) * B(4x16) + C
  return __builtin_amdgcn_wmma_f32_16x16x4_f32(false, a, false, b, (short)0, c,
                                               false, false);
}

// ---- CDNA5 async global->LDS copies (ASYNCcnt-tracked) ---------------------
__device__ __forceinline__ void async_b32(unsigned lds_off, const float* g) {
  asm volatile("global_load_async_to_lds_b32 %0, %1, off"
               :: "v"(lds_off), "v"(g) : "memory");
}
__device__ __forceinline__ void async_b128(unsigned lds_off, const float* g) {
  asm volatile("global_load_async_to_lds_b128 %0, %1, off"
               :: "v"(lds_off), "v"(g) : "memory");
}
__device__ __forceinline__ void wait_async0() {
  asm volatile("s_wait_asynccnt 0x0" ::: "memory");
}

// ---------------------------------------------------------------------------
// Kernel 0: zero the accumulation region of the workspace (graph-replay safe)
// ---------------------------------------------------------------------------
__global__ void zero_ws_kernel(float* __restrict__ ws) {
  int i = blockIdx.x * 256 + threadIdx.x;
  if (i < WS_ZERO) ws[i] = 0.0f;
}

// ---------------------------------------------------------------------------
// Kernel 1: channel sums + Gram of the stride-9 subsampled matrix via WMMA.
// Block-cooperative: stage 64ch x 64 subcols in LDS (async gather), each of
// the 8 waves owns exactly 2 of the 16 Gram tiles -> no register spills.
// LDS stride 68: fragment reads are bank-conflict-free (banks = (4m+s) mod 64;
// the two half-waves differ by 2 mod 4, the .x/.y components by 1 mod 4).
// ---------------------------------------------------------------------------
#define COV_STRIDE 68

__global__ void cov_kernel(const float* __restrict__ x, float* __restrict__ ws) {
  __shared__ float Xs[64 * COV_STRIDE];

  const int tid   = threadIdx.x;        // 256 threads = 8 waves
  const int lane  = tid & 31;
  const int wave  = tid >> 5;
  const int r     = wave >> 1;          // A-fragment channel group (0..3)
  const int c0    = (wave & 1) * 2;     // two B-fragment groups: c0, c0+1
  const int m     = lane & 15;
  const int khalf = (lane >> 4) * 2;

  const unsigned lds_base = (unsigned)(size_t)&Xs[0];

  v8f accA = {};   // tile (r, c0)
  v8f accB = {};   // tile (r, c0+1)
  float csum = 0.0f;

  const int nblk = (NSUB + 63) >> 6;    // 911 column-blocks of 64 subcols
  for (int sb = blockIdx.x; sb < nblk; sb += gridDim.x) {
    const int T0 = sb * 64;

    __syncthreads();                    // previous iteration's LDS reads done
    // async gather: Xs[ch][s] = x1[ch, 9*(T0+s)]
    for (int i = tid; i < 4096; i += 256) {
      const int ch = i >> 6;
      const int s  = i & 63;
      int t = T0 + s;
      if (t >= NSUB) t = NSUB - 1;      // clamp; zeroed below
      const int col = 9 * t;
      const float* g = x + (size_t)(col >> 14) * CHW + (size_t)ch * HW +
                       (col & (HW - 1));
      async_b32(lds_base + (unsigned)(ch * COV_STRIDE + s) * 4u, g);
    }
    wait_async0();
    __syncthreads();

    if (T0 + 64 > NSUB) {               // tail cleanup (last block only)
      for (int i = tid; i < 4096; i += 256) {
        const int ch = i >> 6;
        const int s  = i & 63;
        if (T0 + s >= NSUB) Xs[ch * COV_STRIDE + s] = 0.0f;
      }
      __syncthreads();
    }

    // channel sums (threads 0..63 each own one channel row)
    if (tid < 64) {
      float s = 0.0f;
      for (int j = 0; j < 64; ++j) s += Xs[tid * COV_STRIDE + j];
      csum += s;
    }

    // 16 K-steps of K=4 over the staged 64 subcols
    for (int kk = 0; kk < 16; ++kk) {
      const int s0 = kk * 4 + khalf;
      v2f fa, fb0, fb1;
      const int ra  = (r * 16 + m) * COV_STRIDE;
      const int rb0 = (c0 * 16 + m) * COV_STRIDE;
      const int rb1 = ((c0 + 1) * 16 + m) * COV_STRIDE;
      fa.x  = Xs[ra + s0];   fa.y  = Xs[ra + s0 + 1];
      fb0.x = Xs[rb0 + s0];  fb0.y = Xs[rb0 + s0 + 1];
      fb1.x = Xs[rb1 + s0];  fb1.y = Xs[rb1 + s0 + 1];
      accA = wmma_f32x4(fa, fb0, accA);
      accB = wmma_f32x4(fa, fb1, accB);
    }
  }

  // one global atomic per Gram element per block (tiles are wave-disjoint)
  const int mrow = (lane >> 4) * 8;
  const int ncol = lane & 15;
  for (int v = 0; v < 8; ++v) {
    const int row = r * 16 + mrow + v;
    atomicAdd(&ws[WS_G + row * 64 + c0 * 16 + ncol], accA[v]);
    atomicAdd(&ws[WS_G + row * 64 + (c0 + 1) * 16 + ncol], accB[v]);
  }
  if (tid < 64) atomicAdd(&ws[WS_SUM + tid], csum);
}

// ---------------------------------------------------------------------------
// Kernel 2: Newton-Schulz inverse sqrt of cov (single workgroup, LDS-resident)
// ---------------------------------------------------------------------------
__device__ __forceinline__ void mm64(const float* __restrict__ A,
                                     const float* __restrict__ B,
                                     float* __restrict__ C, int tid) {
  const int i  = tid >> 2;
  const int j0 = (tid & 3) * 16;
  float acc[16];
  for (int j = 0; j < 16; ++j) acc[j] = 0.0f;
  for (int k = 0; k < 64; ++k) {
    const float a = A[i * 64 + k];
    for (int j = 0; j < 16; ++j) acc[j] = fmaf(a, B[k * 64 + j0 + j], acc[j]);
  }
  for (int j = 0; j < 16; ++j) C[i * 64 + j0 + j] = acc[j];
}

__global__ void newton_schulz_kernel(float* __restrict__ ws) {
  __shared__ float Y[4096];
  __shared__ float Z[4096];
  __shared__ float T[4096];
  __shared__ float W[4096];
  __shared__ float meanv[64];
  __shared__ float red[256];

  const int tid = threadIdx.x;
  const float inv_n = 1.0f / (float)NSUB;

  if (tid < 64) meanv[tid] = ws[WS_SUM + tid] * inv_n;

  for (int i = tid; i < 4096; i += 256) {
    float v = ws[WS_G + i] * inv_n;
    if ((i >> 6) == (i & 63)) v += EPSI;
    Y[i] = v;                            // A = G/n + eps*I
  }
  __syncthreads();

  float s = 0.0f;
  for (int i = tid; i < 4096; i += 256) s += Y[i] * Y[i];
  red[tid] = s;
  __syncthreads();
  for (int off = 128; off > 0; off >>= 1) {
    if (tid < off) red[tid] += red[tid + off];
    __syncthreads();
  }
  const float normA = sqrtf(red[0]);
  const float inv_norm = 1.0f / normA;

  for (int i = tid; i < 4096; i += 256) {
    Y[i] *= inv_norm;
    Z[i] = ((i >> 6) == (i & 63)) ? 1.0f : 0.0f;
  }
  __syncthreads();

  for (int it = 0; it < 5; ++it) {
    mm64(Z, Y, W, tid);                  // W = Z@Y
    __syncthreads();
    for (int i = tid; i < 4096; i += 256) {
      const float d = ((i >> 6) == (i & 63)) ? 1.5f : 0.0f;
      T[i] = d - 0.5f * W[i];            // T = 0.5*(3I - Z@Y)
    }
    __syncthreads();
    mm64(Y, T, W, tid);                  // W = Ynew
    __syncthreads();
    mm64(T, Z, Y, tid);                  // Y(buf) = Znew
    __syncthreads();
    for (int i = tid; i < 4096; i += 256) {
      Z[i] = Y[i];
      Y[i] = W[i];
    }
    __syncthreads();
  }

  const float sc = rsqrtf(normA);
  for (int i = tid; i < 4096; i += 256) ws[WS_D + i] = Z[i] * sc;
  if (tid < 64) {
    float b = 0.0f;
    for (int k = 0; k < 64; ++k) b = fmaf(Z[tid * 64 + k] * sc, meanv[k], b);
    ws[WS_BIAS + tid] = -b;
  }
}

// ---------------------------------------------------------------------------
// Kernel 3: y = D @ x1 + bias via WMMA f32, double-buffered async staging.
// Block: 8 waves; wave w: row-tile (w&3), column-half (w>>2) of a 128-col tile.
// ---------------------------------------------------------------------------
#define LDS_STRIDE 136   // 128 + 8 pad: half-waves (k vs k+2 rows) on disjoint banks

__device__ __forceinline__ void stage_tile_async(const float* __restrict__ x,
                                                 int cb, unsigned lbase, int tid) {
  const int J0  = cb * 128;
  const int n   = J0 >> 14;
  const int hw0 = J0 & (HW - 1);
  const float* __restrict__ src = x + (size_t)n * CHW + hw0;
  for (int i = tid; i < 64 * 32; i += 256) {     // 2048 x float4
    const int kr = i >> 5;
    const int jc = (i & 31) * 4;
    async_b128(lbase + (unsigned)(kr * LDS_STRIDE + jc) * 4u,
               src + (size_t)kr * HW + jc);
  }
}

__global__ void gemm_apply_kernel(const float* __restrict__ x,
                                  const float* __restrict__ ws,
                                  float* __restrict__ out) {
  __shared__ float xt[2][64 * LDS_STRIDE];

  const float* __restrict__ Dm   = ws + WS_D;
  const float* __restrict__ bias = ws + WS_BIAS;

  const int tid   = threadIdx.x;
  const int lane  = tid & 31;
  const int wave  = tid >> 5;
  const int ch0   = (wave & 3) * 16;
  const int jbase = (wave >> 2) * 64;
  const int m     = lane & 15;
  const int khalf = (lane >> 4) * 2;

  const unsigned lbase0 = (unsigned)(size_t)&xt[0][0];
  const unsigned lbase1 = (unsigned)(size_t)&xt[1][0];

  // Preload all A fragments for this row-tile (deconv matrix, constant)
  v2f afrag[16];
  for (int kk = 0; kk < 16; ++kk) {
    const int k = kk * 4 + khalf;
    afrag[kk].x = Dm[(ch0 + m) * 64 + k];
    afrag[kk].y = Dm[(ch0 + m) * 64 + k + 1];
  }
  const int mrow_off = (lane >> 4) * 8;
  float bl[8];
  for (int v = 0; v < 8; ++v) bl[v] = bias[ch0 + mrow_off + v];

  const int numColBlocks = NCOLS / 128;          // 4096
  const int stride = gridDim.x;

  // prologue: fill buffer 0
  stage_tile_async(x, blockIdx.x, lbase0, tid);
  wait_async0();
  __syncthreads();

  int it = 0;
  for (int cb = blockIdx.x; cb < numColBlocks; cb += stride, ++it) {
    const int cur = it & 1;
    const float* __restrict__ xcur = &xt[cur][0];

    // prefetch next tile into the other buffer (overlaps with compute)
    const int cbn = cb + stride;
    if (cbn < numColBlocks)
      stage_tile_async(x, cbn, cur ? lbase0 : lbase1, tid);

    v8f acc0 = {}, acc1 = {}, acc2 = {}, acc3 = {};
    const int jl = jbase + m;
    for (int kk = 0; kk < 16; ++kk) {
      const int krow = kk * 4 + khalf;
      const float* __restrict__ r0 = &xcur[krow * LDS_STRIDE];
      const float* __restrict__ r1 = &xcur[(krow + 1) * LDS_STRIDE];
      v2f b0; b0.x = r0[jl];      b0.y = r1[jl];
      v2f b1; b1.x = r0[jl + 16]; b1.y = r1[jl + 16];
      v2f b2; b2.x = r0[jl + 32]; b2.y = r1[jl + 32];
      v2f b3; b3.x = r0[jl + 48]; b3.y = r1[jl + 48];
      acc0 = wmma_f32x4(afrag[kk], b0, acc0);
      acc1 = wmma_f32x4(afrag[kk], b1, acc1);
      acc2 = wmma_f32x4(afrag[kk], b2, acc2);
      acc3 = wmma_f32x4(afrag[kk], b3, acc3);
    }

    // store: lane holds rows ch0+mrow_off+v, cols jbase+m+{0,16,32,48}
    const int J0  = cb * 128;
    const int n   = J0 >> 14;
    const int hw0 = J0 & (HW - 1);
    float* __restrict__ dstp = out + (size_t)n * CHW + hw0 + jbase + m;
    for (int v = 0; v < 8; ++v) {
      const int ch = ch0 + mrow_off + v;
      float* __restrict__ drow = dstp + (size_t)ch * HW;
      drow[0]  = acc0[v] + bl[v];
      drow[16] = acc1[v] + bl[v];
      drow[32] = acc2[v] + bl[v];
      drow[48] = acc3[v] + bl[v];
    }

    wait_async0();       // next buffer filled (this wave's copies)
    __syncthreads();     // all waves done reading cur + done staging
  }
}

// ---------------------------------------------------------------------------
extern "C" void kernel_launch(void* const* d_in, const int* in_sizes, int n_in,
                              void* d_out, int out_size, void* d_ws, size_t ws_size,
                              hipStream_t stream) {
  const float* x = (const float*)d_in[0];
  float* out = (float*)d_out;
  float* ws  = (float*)d_ws;

  zero_ws_kernel<<<(WS_ZERO + 255) / 256, 256, 0, stream>>>(ws);
  cov_kernel<<<128, 256, 0, stream>>>(x, ws);
  newton_schulz_kernel<<<1, 256, 0, stream>>>(ws);
  gemm_apply_kernel<<<1024, 256, 0, stream>>>(x, ws, out);
}